// LocalAttender_88957362635228
// MI455X (gfx1250) — compile-verified
//
#include <hip/hip_runtime.h>

typedef __attribute__((ext_vector_type(2))) float v2f;
typedef __attribute__((ext_vector_type(8))) float v8f;

#define B_      8
#define C_      128
#define Hv      64
#define Wv      64
#define Ho      128
#define Wo      128
#define Kc      9
#define TILE    16            // 16x16 output pixels per workgroup
#define VH      10            // value tile 8x8 + halo 1
#define GPLANE  (Ho * Wo)     // 16384
#define GBATCH  (C_ * GPLANE)
#define VPLANE  (Hv * Wv)

__global__ __launch_bounds__(256) void local_attender_kernel(
    const float* __restrict__ guide, const float* __restrict__ value,
    const float* __restrict__ w_conv, const float* __restrict__ b_conv,
    float* __restrict__ out)
{
  __shared__ float wT[C_ * 16];           // [c][k], k zero-padded to 16 (8 KB)
  __shared__ float vT[C_ * VH * VH];      // [c][vy][vx] value tile + halo (51.2 KB)
  __shared__ float attL[TILE * TILE * 17];// [pixel][k], stride 17 = conflict-free (17.4 KB)

  const int tid = threadIdx.x;
  const int bz  = blockIdx.z;
  const int oy0 = blockIdx.y * TILE;
  const int ox0 = blockIdx.x * TILE;
  const int vy0 = oy0 >> 1;
  const int vx0 = ox0 >> 1;

  // ---- phase 0: stage weights (transposed, padded) and value tile in LDS ----
  for (int i = tid; i < C_ * 16; i += 256) {
    const int c = i >> 4, n = i & 15;
    wT[i] = (n < Kc) ? w_conv[n * C_ + c] : 0.0f;
  }
  for (int i = tid; i < C_ * VH * VH; i += 256) {
    const int c  = i / (VH * VH);
    const int p  = i - c * (VH * VH);
    const int vy = p / VH, vx = p - vy * VH;
    const int gh = min(max(vy0 - 1 + vy, 0), Hv - 1);   // replication padding
    const int gw = min(max(vx0 - 1 + vx, 0), Wv - 1);
    vT[i] = value[(size_t)(bz * C_ + c) * VPLANE + gh * Wv + gw];
  }
  __syncthreads();

  // ---- phase 1: att_linear[pix][k] = sum_c guide[c,pix] * w[k,c] via WMMA f32 16x16x4 ----
  const int lane = tid & 31;
  const int wave = tid >> 5;
  const int n    = lane & 15;     // matrix M (pixel-x) for A, N (k index) for B
  const int hi   = lane >> 4;     // 0 -> K pair {0,1}; 1 -> K pair {2,3}

  for (int rep = 0; rep < 2; ++rep) {
    const int gy = oy0 + wave * 2 + rep;
    const float* gbase = guide + (size_t)bz * GBATCH + (size_t)gy * Wo + ox0 + n;
    v8f acc = {0.f, 0.f, 0.f, 0.f, 0.f, 0.f, 0.f, 0.f};
    for (int c0 = 0; c0 < C_; c0 += 4) {
      const int cA = c0 + (hi << 1);
      v2f a, b;
      a.x = gbase[(size_t)cA * GPLANE];        // A[M=n, K=cA]
      a.y = gbase[(size_t)(cA + 1) * GPLANE];  // A[M=n, K=cA+1]
      b.x = wT[cA * 16 + n];                   // B[K=cA, N=n]
      b.y = wT[(cA + 1) * 16 + n];             // B[K=cA+1, N=n]
      acc = __builtin_amdgcn_wmma_f32_16x16x4_f32(
          false, a, false, b, (short)0, acc, false, false);
    }
    const int row = wave * 2 + rep;
#pragma unroll
    for (int r = 0; r < 8; ++r) {              // D: vgpr r, lanes<16 -> M=r, else M=r+8
      const int txD = r + (hi << 3);
      attL[(row * TILE + txD) * 17 + n] = acc[r];
    }
  }
  __syncthreads();

  // ---- phase 2: residual + bias + softmax(9), then 3x3 weighted gather over 128 ch ----
  {
    const int p  = tid;
    const int ty = p >> 4, tx = p & 15;
    const int gy = oy0 + ty, gx = ox0 + tx;
    const size_t gpix = (size_t)bz * GBATCH + (size_t)gy * Wo + gx;

    float a[Kc];
    float mx = -3.4e38f;
#pragma unroll
    for (int k = 0; k < Kc; ++k) {
      a[k] = attL[p * 17 + k] + guide[gpix + (size_t)k * GPLANE] + b_conv[k];
      mx = fmaxf(mx, a[k]);
    }
    float s = 0.f;
#pragma unroll
    for (int k = 0; k < Kc; ++k) { a[k] = __expf(a[k] - mx); s += a[k]; }
    const float inv = 1.0f / s;
#pragma unroll
    for (int k = 0; k < Kc; ++k) a[k] *= inv;

    const int vly = (ty >> 1) + 1, vlx = (tx >> 1) + 1;
    const float* vb = vT + (vly - 1) * VH + (vlx - 1);
#pragma unroll 4
    for (int c = 0; c < C_; ++c) {
      const float* vc = vb + c * (VH * VH);
      float r = a[0] * vc[0]          + a[1] * vc[1]          + a[2] * vc[2]
              + a[3] * vc[VH]         + a[4] * vc[VH + 1]     + a[5] * vc[VH + 2]
              + a[6] * vc[2 * VH]     + a[7] * vc[2 * VH + 1] + a[8] * vc[2 * VH + 2];
      out[gpix + (size_t)c * GPLANE] = r;
    }
  }
}

extern "C" void kernel_launch(void* const* d_in, const int* in_sizes, int n_in,
                              void* d_out, int out_size, void* d_ws, size_t ws_size,
                              hipStream_t stream) {
  const float* guide  = (const float*)d_in[0];
  const float* value  = (const float*)d_in[1];
  const float* w_conv = (const float*)d_in[2];
  const float* b_conv = (const float*)d_in[3];
  float* out = (float*)d_out;

  dim3 grid(Wo / TILE, Ho / TILE, B_);  // 8 x 8 x 8
  dim3 block(256);                       // 8 wave32
  local_attender_kernel<<<grid, block, 0, stream>>>(guide, value, w_conv, b_conv, out);
}